// AngleClassificationLoss_12970801234035
// MI455X (gfx1250) — compile-verified
//
#include <hip/hip_runtime.h>
#include <hip/hip_bf16.h>
#include <stdint.h>

// ---------------------------------------------------------------------------
// AngleClassificationLoss for MI455X (gfx1250, wave32)
//   loss = [ sum_all( -max(log1p(-p),-100) ) + sum_patch( y*(lq-lp) ) ] / (B*180*360)
// HBM-bound: 531 MB streamed once -> ~23 us floor at 23.3 TB/s.
// Streaming uses global_load_async_to_lds_b128 (ASYNCcnt pipeline, depth 4);
// wave reduction uses v_wmma_f32_16x16x4_f32 with a ones A-matrix.
// ---------------------------------------------------------------------------

#define N_THETA 180
#define N_PHI   360
#define BATCH   1024

constexpr int       kThreads     = 256;                 // 8 waves
constexpr int       kBulkBlocks  = 1620;
constexpr int       kBulkIters   = 80;                  // 1620*256*80 == 33,177,600 float4
constexpr int       kDepth       = 4;                   // async pipeline depth (16 KB LDS)
constexpr long long kTotalF4     = 33177600LL;          // 2*B*180*360 / 4
constexpr int       kPatchBlocks = 2 * BATCH;           // one per (batch, channel)
constexpr int       kNPartials   = kBulkBlocks + kPatchBlocks;
constexpr int       kRad         = 15;                  // exp(-d^2/2) == 0 in f32 beyond ~14

typedef __attribute__((ext_vector_type(2))) float v2f;
typedef __attribute__((ext_vector_type(8))) float v8f;
typedef int v4i __attribute__((__vector_size__(16)));   // matches builtin's V4i

#define AS1 __attribute__((address_space(1)))
#define AS3 __attribute__((address_space(3)))

__device__ __forceinline__ AS3 void* to_lds(void* p) {
  // Generic LDS address: low 32 bits ARE the LDS offset (aperture rule).
  return (AS3 void*)(uint32_t)(uintptr_t)p;
}

template <int N>
__device__ __forceinline__ void wait_async() {
#if __has_builtin(__builtin_amdgcn_s_wait_asynccnt)
  __builtin_amdgcn_s_wait_asynccnt((short)N);
#else
  asm volatile("s_wait_asynccnt %0" :: "n"(N) : "memory");
#endif
}

__device__ __forceinline__ void async_copy_f4(const float4* gp, float4* lp) {
#if __has_builtin(__builtin_amdgcn_global_load_async_to_lds_b128)
  __builtin_amdgcn_global_load_async_to_lds_b128(
      (AS1 v4i*)(uintptr_t)gp, (AS3 v4i*)to_lds(lp), 0, 0);
#else
  unsigned lds_off = (unsigned)(uintptr_t)to_lds(lp);
  asm volatile("global_load_async_to_lds_b128 %0, %1, off"
               :: "v"(lds_off), "v"(gp) : "memory");
#endif
}

// -------------------------- bulk streaming kernel --------------------------
__global__ void __launch_bounds__(kThreads)
bulk_logq_kernel(const float4* __restrict__ probs, float* __restrict__ partials) {
  __shared__ float4 stage[kDepth][kThreads];
  __shared__ float  wavesum[kThreads / 32];

  const int t = threadIdx.x;
  const long long base = (long long)blockIdx.x * (long long)(kThreads * kBulkIters);

  auto issue = [&](int s) {
    long long idx = base + (long long)s * kThreads + t;
    idx = (idx < kTotalF4 - 1) ? idx : (kTotalF4 - 1);   // branchless clamp (prologue overshoot)
    async_copy_f4(probs + idx, &stage[s & (kDepth - 1)][t]);
  };

  issue(0); issue(1); issue(2);

  float acc = 0.0f;
  for (int s = 0; s < kBulkIters; ++s) {
    issue(s + (kDepth - 1));
    wait_async<kDepth - 1>();               // stage s complete (in-order async loads)
    float4 v = stage[s & (kDepth - 1)][t];  // each lane reads only bytes it staged: no barrier
    acc += fminf(-log1pf(-v.x), 100.0f);
    acc += fminf(-log1pf(-v.y), 100.0f);
    acc += fminf(-log1pf(-v.z), 100.0f);
    acc += fminf(-log1pf(-v.w), 100.0f);
  }
  wait_async<0>();                          // drain clamped overshoot before LDS reuse / exit

  // Wave reduction: WMMA pair-sum (K=4, ones A-matrix) then 16-lane xor tree.
  float r = acc;
#if __has_builtin(__builtin_amdgcn_wmma_f32_16x16x4_f32)
  {
    v2f a;  a[0] = 1.0f; a[1] = 1.0f;       // A = ones(16x4)
    v2f bm; bm[0] = r;  bm[1] = 0.0f;       // B(4x16): lane accumulators on K=0/K=2 rows
    v8f cz = {};
    v8f d = __builtin_amdgcn_wmma_f32_16x16x4_f32(false, a, false, bm, (short)0, cz,
                                                  false, false);
    r = d[0];                               // lane L: acc[L%16] + acc[L%16+16]
    r += __shfl_xor(r, 8, 32);
    r += __shfl_xor(r, 4, 32);
    r += __shfl_xor(r, 2, 32);
    r += __shfl_xor(r, 1, 32);
  }
#else
  for (int m = 16; m; m >>= 1) r += __shfl_xor(r, m, 32);
#endif

  if ((t & 31) == 0) wavesum[t >> 5] = r;
  __syncthreads();
  if (t == 0) {
    float s = 0.0f;
    for (int w = 0; w < kThreads / 32; ++w) s += wavesum[w];
    partials[blockIdx.x] = s;
  }
}

// -------------------------- sparse label correction ------------------------
__device__ __forceinline__ float block_sum_bcast(float v, volatile float* red) {
  for (int m = 16; m; m >>= 1) v += __shfl_xor(v, m, 32);
  const int t = threadIdx.x, lane = t & 31, wid = t >> 5;
  if (lane == 0) red[wid] = v;
  __syncthreads();
  if (t == 0) {
    float s = 0.0f;
    for (int w = 0; w < kThreads / 32; ++w) s += red[w];
    red[0] = s;
  }
  __syncthreads();
  float out = red[0];
  __syncthreads();
  return out;
}

__global__ void __launch_bounds__(kThreads)
patch_kernel(const float* __restrict__ probs, const float* __restrict__ gt_u,
             const float* __restrict__ gt_r, float* __restrict__ partials) {
  __shared__ float red[kThreads / 32];

  constexpr float kPi    = 3.14159265358979323846f;
  constexpr float kTwoPi = 6.28318530717958647692f;

  const int bc = blockIdx.x;
  const int b  = bc >> 1;
  const int ch = bc & 1;
  const float* g = (ch == 0 ? gt_u : gt_r) + 3 * b;

  const float vz    = fminf(fmaxf(g[2], -1.0f), 1.0f);
  const float theta = acosf(vz);
  float phi = atan2f(g[1], g[0]);
  phi = (phi < 0.0f) ? (phi + kTwoPi) : phi;

  int ti = (int)(theta / kPi * (float)N_THETA);  ti = min(max(ti, 0), N_THETA - 1);
  int pj = (int)(phi / kTwoPi * (float)N_PHI);   pj = min(max(pj, 0), N_PHI - 1);

  const int i0 = max(ti - kRad, 0), i1 = min(ti + kRad, N_THETA - 1);
  const int j0 = max(pj - kRad, 0), j1 = min(pj + kRad, N_PHI - 1);
  const int ni = i1 - i0 + 1, nj = j1 - j0 + 1, n = ni * nj;

  // Pass 1: normalization (patch weight sum == full masked kernel sum in f32).
  float wloc = 0.0f;
  for (int k = threadIdx.x; k < n; k += kThreads) {
    const int di = i0 + k / nj - ti;
    const int dj = j0 + k % nj - pj;
    wloc += expf(-0.5f * (float)(di * di + dj * dj));
  }
  const float W    = block_sum_bcast(wloc, red);
  const float invW = 1.0f / W;

  // Pass 2: sum y*(lq - lp) over the patch.
  const float* basep = probs + (size_t)(b * 2 + ch) * (N_THETA * N_PHI);
  float csum = 0.0f;
  for (int k = threadIdx.x; k < n; k += kThreads) {
    const int i  = i0 + k / nj, j = j0 + k % nj;
    const int di = i - ti,      dj = j - pj;
    const float y  = expf(-0.5f * (float)(di * di + dj * dj)) * invW;
    const float p  = basep[i * N_PHI + j];
    const float lp = fmaxf(logf(p),     -100.0f);
    const float lq = fmaxf(log1pf(-p),  -100.0f);
    csum += y * (lq - lp);
  }
  const float C = block_sum_bcast(csum, red);
  if (threadIdx.x == 0) partials[bc] = C;
}

// -------------------------- deterministic finalize -------------------------
__global__ void __launch_bounds__(kThreads)
finalize_kernel(const float* __restrict__ partials, float* __restrict__ out) {
  __shared__ double red[kThreads / 32];
  double acc = 0.0;
  for (int k = threadIdx.x; k < kNPartials; k += kThreads) acc += (double)partials[k];
  for (int m = 16; m; m >>= 1) acc += __shfl_xor(acc, m, 32);
  const int t = threadIdx.x, lane = t & 31, wid = t >> 5;
  if (lane == 0) red[wid] = acc;
  __syncthreads();
  if (t == 0) {
    double s = 0.0;
    for (int w = 0; w < kThreads / 32; ++w) s += red[w];
    out[0] = (float)(s / (double)((long long)BATCH * N_THETA * N_PHI));
  }
}

// ---------------------------------------------------------------------------
extern "C" void kernel_launch(void* const* d_in, const int* in_sizes, int n_in,
                              void* d_out, int out_size, void* d_ws, size_t ws_size,
                              hipStream_t stream) {
  (void)in_sizes; (void)n_in; (void)out_size; (void)ws_size;
  const float* probs = (const float*)d_in[0];
  const float* gt_u  = (const float*)d_in[1];
  const float* gt_r  = (const float*)d_in[2];
  float* part = (float*)d_ws;  // kNPartials floats (~14.7 KB)

  bulk_logq_kernel<<<kBulkBlocks, kThreads, 0, stream>>>((const float4*)probs, part);
  patch_kernel<<<kPatchBlocks, kThreads, 0, stream>>>(probs, gt_u, gt_r,
                                                      part + kBulkBlocks);
  finalize_kernel<<<1, kThreads, 0, stream>>>(part, (float*)d_out);
}